// MeanResidualLowRankMixEnsemble_47227460387331
// MI455X (gfx1250) — compile-verified
//
#include <hip/hip_runtime.h>

typedef float v2f __attribute__((ext_vector_type(2)));
typedef float v8f __attribute__((ext_vector_type(8)));

#define B_ 512
#define M_ 3
#define L_ 50000
#define A_ 20000
#define R_ 64
#define KSPLIT 20
#define KCHUNK (A_ / KSPLIT) /* 1000, divisible by 4 */

// workspace layout (float offsets)
#define WS_W 0          // 3 floats: softmax weights
#define WS_ALPHA 3      // 1 float
#define WS_S 4          // 64 floats: column sums of V
#define WS_AM 128       // 512 floats: alpha * mean[b]
#define WS_T 1024       // 512*64 = 32768 floats: T = p @ U
#define WS_TPART 65536  // KSPLIT * 32768 floats: split-K partials
// total: 720896 floats = 2.88 MB

// fast sigmoid: v_exp_f32 + v_rcp_f32 (~1 ulp rcp; fine under tolerance)
__device__ __forceinline__ float sigmoidf(float x) {
    return __builtin_amdgcn_rcpf(1.0f + __expf(-x));
}

__global__ void k_prep(const float* __restrict__ gl, const float* __restrict__ la,
                       float* __restrict__ ws) {
    if (threadIdx.x == 0) {
        float g0 = gl[0], g1 = gl[1], g2 = gl[2];
        float mx = fmaxf(g0, fmaxf(g1, g2));
        float e0 = __expf(g0 - mx), e1 = __expf(g1 - mx), e2 = __expf(g2 - mx);
        float inv = 1.0f / (e0 + e1 + e2);
        ws[WS_W + 0] = e0 * inv;
        ws[WS_W + 1] = e1 * inv;
        ws[WS_W + 2] = e2 * inv;
        ws[WS_ALPHA] = 1.0f / (1.0f + __expf(-la[0]));  // ALPHA_MAX = 1.0
    }
}

// base[b,l] = sum_m x[b,m,l] * (w[m] + delta_w[m,l]) + bias[l]
__global__ void k_base(const float* __restrict__ x, const float* __restrict__ dw,
                       const float* __restrict__ bias, const float* __restrict__ ws,
                       float* __restrict__ out) {
    int t = blockIdx.x * blockDim.x + threadIdx.x;  // B*L/4 threads
    int b = t / (L_ / 4);
    if (b >= B_) return;
    int l = (t % (L_ / 4)) * 4;
    float w0 = ws[WS_W + 0], w1 = ws[WS_W + 1], w2 = ws[WS_W + 2];
    const float* xb = x + (size_t)b * (M_ * L_);
    float4 x0 = *(const float4*)(xb + 0 * L_ + l);
    float4 x1 = *(const float4*)(xb + 1 * L_ + l);
    float4 x2 = *(const float4*)(xb + 2 * L_ + l);
    float4 d0 = *(const float4*)(dw + 0 * L_ + l);
    float4 d1 = *(const float4*)(dw + 1 * L_ + l);
    float4 d2 = *(const float4*)(dw + 2 * L_ + l);
    float4 bv = *(const float4*)(bias + l);
    float4 r;
    r.x = x0.x * (w0 + d0.x) + x1.x * (w1 + d1.x) + x2.x * (w2 + d2.x) + bv.x;
    r.y = x0.y * (w0 + d0.y) + x1.y * (w1 + d1.y) + x2.y * (w2 + d2.y) + bv.y;
    r.z = x0.z * (w0 + d0.z) + x1.z * (w1 + d1.z) + x2.z * (w2 + d2.z) + bv.z;
    r.w = x0.w * (w0 + d0.w) + x1.w * (w1 + d1.w) + x2.w * (w2 + d2.w) + bv.w;
    *(float4*)(out + (size_t)b * L_ + l) = r;
}

// GEMM1 split-K partials: Tpart[split] += sigmoid(base[:,idx]) @ U  (16x16 tile per wave)
__global__ void k_gemm1(const float* __restrict__ base, const int* __restrict__ idx,
                        const float* __restrict__ U, float* __restrict__ ws) {
    int lane = threadIdx.x;
    int half = lane >> 4;   // K-group select
    int lm = lane & 15;     // M (batch rows) / N (U cols)
    int split = blockIdx.x;
    int b0 = blockIdx.y << 4;
    int n0 = blockIdx.z << 4;
    const float* brow = base + (size_t)(b0 + lm) * L_;
    v8f acc = {};
    int k0 = split * KCHUNK;
#pragma unroll 4
    for (int k = k0; k < k0 + KCHUNK; k += 4) {
        int kk = k + 2 * half;           // even -> 8B-aligned idx pair
        int2 c = *(const int2*)(idx + kk);
        v2f a;
        a.x = sigmoidf(brow[c.x]);
        a.y = sigmoidf(brow[c.y]);
        v2f bf;
        bf.x = U[(size_t)kk * R_ + n0 + lm];
        bf.y = U[(size_t)(kk + 1) * R_ + n0 + lm];
        acc = __builtin_amdgcn_wmma_f32_16x16x4_f32(false, a, false, bf,
                                                    (short)0, acc, false, false);
    }
    float* tp = ws + WS_TPART + (size_t)split * (B_ * R_);
#pragma unroll
    for (int v = 0; v < 8; ++v) {
        int row = b0 + v + 8 * half;
        tp[row * R_ + n0 + lm] = acc[v];
    }
}

__global__ void k_reduceT(float* __restrict__ ws) {
    int i = blockIdx.x * blockDim.x + threadIdx.x;
    if (i >= B_ * R_) return;
    float s = 0.f;
#pragma unroll
    for (int p = 0; p < KSPLIT; ++p) s += ws[WS_TPART + p * (B_ * R_) + i];
    ws[WS_T + i] = s;
}

// S[r] = sum_a V[a,r]
__global__ void k_colsumV(const float* __restrict__ V, float* __restrict__ ws) {
    __shared__ float red[256];
    int r = blockIdx.x;
    float s = 0.f;
    for (int a = threadIdx.x; a < A_; a += 256) s += V[(size_t)a * R_ + r];
    red[threadIdx.x] = s;
    __syncthreads();
    for (int off = 128; off > 0; off >>= 1) {
        if (threadIdx.x < off) red[threadIdx.x] += red[threadIdx.x + off];
        __syncthreads();
    }
    if (threadIdx.x == 0) ws[WS_S + r] = red[0];
}

// am[b] = alpha * (T[b,:] . S) / A   (== alpha * mean_a delta[b,a])
__global__ void k_mean(float* __restrict__ ws) {
    int b = blockIdx.x * blockDim.x + threadIdx.x;
    if (b >= B_) return;
    float alpha = ws[WS_ALPHA];
    float dot = 0.f;
#pragma unroll
    for (int r = 0; r < R_; ++r) dot += ws[WS_T + b * R_ + r] * ws[WS_S + r];
    ws[WS_AM + b] = alpha * dot * (1.0f / (float)A_);
}

// delta tile = T @ V^T (16x16, K=64 fully unrolled), then fused:
// out[b, idx[a]] += alpha*delta[b,a] - alpha*mean[b]
__global__ void k_gemm2_scatter(const float* __restrict__ V, const int* __restrict__ idx,
                                const float* __restrict__ ws, float* __restrict__ out) {
    int lane = threadIdx.x;
    int half = lane >> 4;
    int lm = lane & 15;
    int a0 = blockIdx.x << 4;
    int b0 = blockIdx.y << 4;
    const float* trow = ws + WS_T + (size_t)(b0 + lm) * R_;
    const float* vrow = V + (size_t)(a0 + lm) * R_;
    float alpha = ws[WS_ALPHA];
    v8f acc = {};
#pragma unroll
    for (int k = 0; k < R_; k += 4) {
        int kk = k + 2 * half;  // even -> 8B-aligned float2 pairs
        v2f a = *(const v2f*)(trow + kk);
        v2f bf = *(const v2f*)(vrow + kk);
        acc = __builtin_amdgcn_wmma_f32_16x16x4_f32(false, a, false, bf,
                                                    (short)0, acc, false, false);
    }
    int l = idx[a0 + lm];  // scatter column for this lane
#pragma unroll
    for (int v = 0; v < 8; ++v) {
        int row = b0 + v + 8 * half;
        float val = alpha * acc[v] - ws[WS_AM + row];
        unsafeAtomicAdd(out + (size_t)row * L_ + l, val);
    }
}

extern "C" void kernel_launch(void* const* d_in, const int* in_sizes, int n_in,
                              void* d_out, int out_size, void* d_ws, size_t ws_size,
                              hipStream_t stream) {
    const float* x    = (const float*)d_in[0];  // (B,M,L)
    const int*   idx  = (const int*)d_in[1];    // (A,)
    const float* gl   = (const float*)d_in[2];  // (M,)
    const float* dw   = (const float*)d_in[3];  // (M,L)
    const float* bias = (const float*)d_in[4];  // (L,)
    const float* la   = (const float*)d_in[5];  // scalar
    const float* U    = (const float*)d_in[6];  // (A,R)
    const float* V    = (const float*)d_in[7];  // (A,R)
    float* out = (float*)d_out;                 // (B,L)
    float* ws  = (float*)d_ws;

    k_prep<<<1, 32, 0, stream>>>(gl, la, ws);
    k_base<<<(B_ * L_ / 4 + 255) / 256, 256, 0, stream>>>(x, dw, bias, ws, out);
    dim3 g1(KSPLIT, B_ / 16, R_ / 16);
    k_gemm1<<<g1, 32, 0, stream>>>(out, idx, U, ws);
    k_reduceT<<<(B_ * R_ + 255) / 256, 256, 0, stream>>>(ws);
    k_colsumV<<<R_, 256, 0, stream>>>(V, ws);
    k_mean<<<(B_ + 255) / 256, 256, 0, stream>>>(ws);
    dim3 g2(A_ / 16, B_ / 16);
    k_gemm2_scatter<<<g2, 32, 0, stream>>>(V, idx, ws, out);
}